// GAE_81398220194430
// MI455X (gfx1250) — compile-verified
//
#include <hip/hip_runtime.h>
#include <hip/hip_bf16.h>

#define N_NODES 16384
#define N_EDGES 524288
#define INC     128
#define H1C     128
#define H2C     64
#define EDGEF   16
#define OUTF    30   // 10 binary + 4 + 6 softmax + 10 continuous

typedef float v2f __attribute__((ext_vector_type(2)));
typedef float v8f __attribute__((ext_vector_type(8)));

// ---------------------------------------------------------------------------
// Degree / normalization
// ---------------------------------------------------------------------------
__global__ void fill_deg(float* deg) {
    int i = blockIdx.x * blockDim.x + threadIdx.x;
    if (i < N_NODES) deg[i] = 1.0f;                 // self loop
}

__global__ void deg_accum(const long long* __restrict__ dst, float* __restrict__ deg) {
    int e = blockIdx.x * blockDim.x + threadIdx.x;
    if (e < N_EDGES) atomicAdd(&deg[(int)dst[e]], 1.0f);
}

__global__ void compute_norm(const long long* __restrict__ src,
                             const long long* __restrict__ dst,
                             const float* __restrict__ deg,
                             float* __restrict__ norm) {
    int e = blockIdx.x * blockDim.x + threadIdx.x;
    if (e < N_EDGES)
        norm[e] = rsqrtf(deg[(int)src[e]]) * rsqrtf(deg[(int)dst[e]]);
}

// ---------------------------------------------------------------------------
// Generic f32 WMMA GEMM: C[M x Nc] = A[M x K] @ B[K x Nc]
// One wave (32 threads) computes one 16x16 tile via V_WMMA_F32_16X16X4_F32.
// A frag : lanes 0-15 hold row M=lane, K={k,k+1}; lanes 16-31 K={k+2,k+3}.
// B frag : VGPR0 = B[k+kh][n], VGPR1 = B[k+kh+1][n], n = lane&15, kh = (lane>>4)*2
// D      : VGPR r -> row r (lanes 0-15) / row r+8 (lanes 16-31), col = lane&15
// ---------------------------------------------------------------------------
__global__ void gemm_f32_wmma(const float* __restrict__ A, const float* __restrict__ B,
                              float* __restrict__ C, int K, int Nc) {
    const int i0   = blockIdx.x * 16;
    const int j0   = blockIdx.y * 16;
    const int lane = threadIdx.x;
    const int m    = lane & 15;
    const int kh   = (lane >> 4) * 2;

    v8f acc = {};
    for (int k = 0; k < K; k += 4) {
        v2f a = *(const v2f*)(A + (long long)(i0 + m) * K + k + kh);
        const float* bp = B + (long long)(k + kh) * Nc + j0 + m;
        v2f b = { bp[0], bp[Nc] };
        acc = __builtin_amdgcn_wmma_f32_16x16x4_f32(false, a, false, b,
                                                    (short)0, acc, false, false);
    }
    const int rbase = (lane >> 4) * 8;
    const int col   = j0 + m;
#pragma unroll
    for (int r = 0; r < 8; ++r)
        C[(long long)(i0 + rbase + r) * Nc + col] = acc[r];
}

// ---------------------------------------------------------------------------
// adj = Z @ Z^T  (Z is [N, 64]).  B fragment of Z^T == A-style load from Z.
// 2x2 register blocking: one wave computes a 32x32 block (4 tiles, 4
// independent accumulation chains -> WMMA RAW hazards hidden, fragment-load
// traffic halved vs 1 tile/wave).  Non-temporal stores: 1.07 GB streamed
// output must not rinse the 192 MB L2 holding z.
// ---------------------------------------------------------------------------
__global__ void adj_wmma(const float* __restrict__ z, float* __restrict__ adj) {
    const int i0   = blockIdx.x * 32;
    const int j0   = blockIdx.y * 32;
    const int lane = threadIdx.x;
    const int m    = lane & 15;
    const int kh   = (lane >> 4) * 2;

    v8f acc00 = {}, acc01 = {}, acc10 = {}, acc11 = {};
#pragma unroll
    for (int k = 0; k < H2C; k += 4) {
        v2f a0 = *(const v2f*)(z + (size_t)(i0 +      m) * H2C + k + kh);
        v2f a1 = *(const v2f*)(z + (size_t)(i0 + 16 + m) * H2C + k + kh);
        v2f b0 = *(const v2f*)(z + (size_t)(j0 +      m) * H2C + k + kh);
        v2f b1 = *(const v2f*)(z + (size_t)(j0 + 16 + m) * H2C + k + kh);
        acc00 = __builtin_amdgcn_wmma_f32_16x16x4_f32(false, a0, false, b0, (short)0, acc00, false, false);
        acc01 = __builtin_amdgcn_wmma_f32_16x16x4_f32(false, a0, false, b1, (short)0, acc01, false, false);
        acc10 = __builtin_amdgcn_wmma_f32_16x16x4_f32(false, a1, false, b0, (short)0, acc10, false, false);
        acc11 = __builtin_amdgcn_wmma_f32_16x16x4_f32(false, a1, false, b1, (short)0, acc11, false, false);
    }
    const int rbase = (lane >> 4) * 8;
#pragma unroll
    for (int r = 0; r < 8; ++r) {
        size_t row0 = (size_t)(i0 + rbase + r) * N_NODES;
        size_t row1 = (size_t)(i0 + 16 + rbase + r) * N_NODES;
        __builtin_nontemporal_store(acc00[r], adj + row0 + j0 + m);
        __builtin_nontemporal_store(acc01[r], adj + row0 + j0 + 16 + m);
        __builtin_nontemporal_store(acc10[r], adj + row1 + j0 + m);
        __builtin_nontemporal_store(acc11[r], adj + row1 + j0 + 16 + m);
    }
}

// ---------------------------------------------------------------------------
// Edge aggregation: agg[dst] += h[src] * norm[e]   (4 channels / thread)
// ---------------------------------------------------------------------------
__global__ void edge_agg(const long long* __restrict__ src, const long long* __restrict__ dst,
                         const float* __restrict__ norm, const float* __restrict__ h,
                         float* __restrict__ agg, int C) {
    const int c4 = C >> 2;
    long long tid = (long long)blockIdx.x * blockDim.x + threadIdx.x;
    long long e = tid / c4;
    if (e >= N_EDGES) return;
    int c = (int)(tid - e * c4) << 2;
    int s = (int)src[e], d = (int)dst[e];
    float w = norm[e];
    float4 hv = *(const float4*)(h + (long long)s * C + c);
    float* ap = agg + (long long)d * C + c;
    atomicAdd(ap + 0, hv.x * w);
    atomicAdd(ap + 1, hv.y * w);
    atomicAdd(ap + 2, hv.z * w);
    atomicAdd(ap + 3, hv.w * w);
}

// Self-loop contribution (+ layer bias):  agg[i][c] += h[i][c]/deg[i] + b[c]
__global__ void self_agg(const float* __restrict__ h, const float* __restrict__ deg,
                         const float* __restrict__ bias, float* __restrict__ agg,
                         int C, long long total) {
    long long i = (long long)blockIdx.x * blockDim.x + threadIdx.x;
    if (i >= total) return;
    long long node = i / C;
    int c = (int)(i - node * C);
    agg[i] += h[i] / deg[node] + bias[c];
}

// ---------------------------------------------------------------------------
// BatchNorm: per-channel mean/var over N, then affine + ReLU (in place)
// ---------------------------------------------------------------------------
__global__ void bn_stats(const float* __restrict__ h, int C,
                         float* __restrict__ mean, float* __restrict__ rstd) {
    __shared__ float ss[256];
    __shared__ float sq[256];
    const int c = blockIdx.x;
    const int t = threadIdx.x;
    float s = 0.f, q = 0.f;
    for (int i = t; i < N_NODES; i += 256) {
        float v = h[(long long)i * C + c];
        s += v; q += v * v;
    }
    ss[t] = s; sq[t] = q;
    __syncthreads();
    for (int w = 128; w > 0; w >>= 1) {
        if (t < w) { ss[t] += ss[t + w]; sq[t] += sq[t + w]; }
        __syncthreads();
    }
    if (t == 0) {
        float m = ss[0] * (1.0f / N_NODES);
        float v = sq[0] * (1.0f / N_NODES) - m * m;
        mean[c] = m;
        rstd[c] = rsqrtf(v + 1e-5f);
    }
}

__global__ void bn_apply_relu(float* __restrict__ h, const float* __restrict__ mean,
                              const float* __restrict__ rstd, const float* __restrict__ g,
                              const float* __restrict__ b, int C, long long total) {
    long long i = (long long)blockIdx.x * blockDim.x + threadIdx.x;
    if (i >= total) return;
    int c = (int)(i % C);
    float v = g[c] * (h[i] - mean[c]) * rstd[c] + b[c];
    h[i] = v > 0.f ? v : 0.f;
}

// ---------------------------------------------------------------------------
// Output heads: [sigmoid(10) | softmax(4) | softmax(6) | linear(10)]
// ---------------------------------------------------------------------------
__global__ void heads_kernel(const float* __restrict__ z,
                             const float* __restrict__ w_bin, const float* __restrict__ b_bin,
                             const float* __restrict__ w_mc,  const float* __restrict__ b_mc,
                             const float* __restrict__ w_cont,const float* __restrict__ b_cont,
                             float* __restrict__ out) {
    int i = blockIdx.x * blockDim.x + threadIdx.x;
    if (i >= N_NODES) return;
    float zi[H2C];
#pragma unroll
    for (int k = 0; k < H2C; ++k) zi[k] = z[(long long)i * H2C + k];
    float* o = out + (long long)i * OUTF;

    for (int j = 0; j < 10; ++j) {                  // binary: sigmoid
        float s = b_bin[j];
        for (int k = 0; k < H2C; ++k) s += zi[k] * w_bin[k * 10 + j];
        o[j] = 1.0f / (1.0f + __expf(-s));
    }
    float mc[10];
    for (int j = 0; j < 10; ++j) {                  // multi-class logits
        float s = b_mc[j];
        for (int k = 0; k < H2C; ++k) s += zi[k] * w_mc[k * 10 + j];
        mc[j] = s;
    }
    {                                               // softmax group of 4
        float mx = mc[0];
        for (int j = 1; j < 4; ++j) mx = fmaxf(mx, mc[j]);
        float ex[4], sum = 0.f;
        for (int j = 0; j < 4; ++j) { ex[j] = __expf(mc[j] - mx); sum += ex[j]; }
        float inv = 1.0f / sum;
        for (int j = 0; j < 4; ++j) o[10 + j] = ex[j] * inv;
    }
    {                                               // softmax group of 6
        float mx = mc[4];
        for (int j = 5; j < 10; ++j) mx = fmaxf(mx, mc[j]);
        float ex[6], sum = 0.f;
        for (int j = 0; j < 6; ++j) { ex[j] = __expf(mc[4 + j] - mx); sum += ex[j]; }
        float inv = 1.0f / sum;
        for (int j = 0; j < 6; ++j) o[14 + j] = ex[j] * inv;
    }
    for (int j = 0; j < 10; ++j) {                  // continuous
        float s = b_cont[j];
        for (int k = 0; k < H2C; ++k) s += zi[k] * w_cont[k * 10 + j];
        o[20 + j] = s;
    }
}

// ---------------------------------------------------------------------------
// Edge features: [z[src] | z[dst]] @ w_edge + b_edge   (thread per edge,feat)
// ---------------------------------------------------------------------------
__global__ void edge_feat(const float* __restrict__ z, const long long* __restrict__ ei,
                          const float* __restrict__ w_edge, const float* __restrict__ b_edge,
                          float* __restrict__ out) {
    long long tid = (long long)blockIdx.x * blockDim.x + threadIdx.x;
    long long e = tid >> 4;
    if (e >= N_EDGES) return;
    int f = (int)(tid & 15);
    int s = (int)ei[e];
    int d = (int)ei[N_EDGES + e];
    float acc = b_edge[f];
    const float* zs = z + (long long)s * H2C;
    const float* zd = z + (long long)d * H2C;
    for (int k = 0; k < H2C; ++k) acc += zs[k] * w_edge[k * EDGEF + f];
    for (int k = 0; k < H2C; ++k) acc += zd[k] * w_edge[(H2C + k) * EDGEF + f];
    __builtin_nontemporal_store(acc, out + e * EDGEF + f);
}

// ---------------------------------------------------------------------------
// Host launcher
// ---------------------------------------------------------------------------
extern "C" void kernel_launch(void* const* d_in, const int* in_sizes, int n_in,
                              void* d_out, int out_size, void* d_ws, size_t ws_size,
                              hipStream_t stream) {
    const float*     x      = (const float*)d_in[0];
    const long long* ei     = (const long long*)d_in[1];   // [2, E] int64
    const float*     w1     = (const float*)d_in[3];
    const float*     b1     = (const float*)d_in[4];
    const float*     bn1_g  = (const float*)d_in[5];
    const float*     bn1_b  = (const float*)d_in[6];
    const float*     w2     = (const float*)d_in[7];
    const float*     b2     = (const float*)d_in[8];
    const float*     bn2_g  = (const float*)d_in[9];
    const float*     bn2_b  = (const float*)d_in[10];
    const float*     w_bin  = (const float*)d_in[11];
    const float*     b_bin  = (const float*)d_in[12];
    const float*     w_mc   = (const float*)d_in[13];
    const float*     b_mc   = (const float*)d_in[14];
    const float*     w_cont = (const float*)d_in[15];
    const float*     b_cont = (const float*)d_in[16];
    const float*     w_edge = (const float*)d_in[17];
    const float*     b_edge = (const float*)d_in[18];

    const long long* srcI = ei;
    const long long* dstI = ei + N_EDGES;

    // Workspace layout (floats), total ~27 MB
    float* ws   = (float*)d_ws;
    float* deg  = ws;                                   // N
    float* norm = deg  + N_NODES;                       // E
    float* h1   = norm + N_EDGES;                       // N*128
    float* agg1 = h1   + (size_t)N_NODES * H1C;         // N*128
    float* h2   = agg1 + (size_t)N_NODES * H1C;         // N*64
    float* agg2 = h2   + (size_t)N_NODES * H2C;         // N*64  (becomes z)
    float* m1   = agg2 + (size_t)N_NODES * H2C;         // 128
    float* r1   = m1 + H1C;                             // 128
    float* m2   = r1 + H1C;                             // 64
    float* r2   = m2 + H2C;                             // 64

    // Output layout (tuple flattened): rec_x | adj | edge_features
    float* out_rec  = (float*)d_out;
    float* out_adj  = out_rec + (size_t)N_NODES * OUTF;
    float* out_edge = out_adj + (size_t)N_NODES * N_NODES;

    hipMemsetAsync(agg1, 0, (size_t)N_NODES * H1C * sizeof(float), stream);
    hipMemsetAsync(agg2, 0, (size_t)N_NODES * H2C * sizeof(float), stream);

    // --- degree / norm ---
    fill_deg<<<N_NODES / 256, 256, 0, stream>>>(deg);
    deg_accum<<<N_EDGES / 256, 256, 0, stream>>>(dstI, deg);
    compute_norm<<<N_EDGES / 256, 256, 0, stream>>>(srcI, dstI, deg, norm);

    // --- layer 1: h1 = x @ w1 ; aggregate ; BN + ReLU ---
    gemm_f32_wmma<<<dim3(N_NODES / 16, H1C / 16), 32, 0, stream>>>(x, w1, h1, INC, H1C);
    {
        long long total = (long long)N_EDGES * (H1C / 4);
        edge_agg<<<(unsigned)((total + 255) / 256), 256, 0, stream>>>(srcI, dstI, norm, h1, agg1, H1C);
    }
    {
        long long total = (long long)N_NODES * H1C;
        self_agg<<<(unsigned)((total + 255) / 256), 256, 0, stream>>>(h1, deg, b1, agg1, H1C, total);
        bn_stats<<<H1C, 256, 0, stream>>>(agg1, H1C, m1, r1);
        bn_apply_relu<<<(unsigned)((total + 255) / 256), 256, 0, stream>>>(agg1, m1, r1, bn1_g, bn1_b, H1C, total);
    }

    // --- layer 2: h2 = h1n @ w2 ; aggregate ; BN + ReLU -> z (in agg2) ---
    gemm_f32_wmma<<<dim3(N_NODES / 16, H2C / 16), 32, 0, stream>>>(agg1, w2, h2, H1C, H2C);
    {
        long long total = (long long)N_EDGES * (H2C / 4);
        edge_agg<<<(unsigned)((total + 255) / 256), 256, 0, stream>>>(srcI, dstI, norm, h2, agg2, H2C);
    }
    {
        long long total = (long long)N_NODES * H2C;
        self_agg<<<(unsigned)((total + 255) / 256), 256, 0, stream>>>(h2, deg, b2, agg2, H2C, total);
        bn_stats<<<H2C, 256, 0, stream>>>(agg2, H2C, m2, r2);
        bn_apply_relu<<<(unsigned)((total + 255) / 256), 256, 0, stream>>>(agg2, m2, r2, bn2_g, bn2_b, H2C, total);
    }
    const float* z = agg2;

    // --- heads ---
    heads_kernel<<<N_NODES / 256, 256, 0, stream>>>(z, w_bin, b_bin, w_mc, b_mc,
                                                    w_cont, b_cont, out_rec);

    // --- adj = z @ z^T (WMMA 2x2-blocked, NT stores; ~1.07 GB streamed) ---
    adj_wmma<<<dim3(N_NODES / 32, N_NODES / 32), 32, 0, stream>>>(z, out_adj);

    // --- edge features ---
    {
        long long total = (long long)N_EDGES * EDGEF;
        edge_feat<<<(unsigned)((total + 255) / 256), 256, 0, stream>>>(z, ei, w_edge, b_edge, out_edge);
    }
}